// Selective_Scan_12214886990398
// MI455X (gfx1250) — compile-verified
//
#include <hip/hip_runtime.h>
#include <hip/hip_bf16.h>
#include <math.h>

#define D_INNER 2048
#define D_STATE 16
#define DT_RANK 64
#define NPROJ   96      // DT_RANK + 2*D_STATE
#define BATCH   2
#define SEQLEN  2048
#define NC      32      // number of scan chunks
#define CL      64      // chunk length (NC*CL == SEQLEN)

typedef __attribute__((ext_vector_type(2))) float v2f;
typedef __attribute__((ext_vector_type(8))) float v8f;

// -------------------------------------------------------------------------
// Kernel 1: x_dbl[b,c,l] = sum_k W[c,k] * x[b,l,k]   (M=96, N=L, K=2048)
// Cs rows (c>=80) get prompt[b,l,c-80] folded in at store time.
// One wave per 16x16 output tile, f32 WMMA 16x16x4 (exact f32 FMA).
// -------------------------------------------------------------------------
__global__ __launch_bounds__(32)
void proj_wmma_kernel(const float* __restrict__ x,       // (B,L,D_INNER)
                      const float* __restrict__ W,       // (96,D_INNER)
                      const float* __restrict__ prompt,  // (B,L,16)
                      float* __restrict__ xdbl)          // (B,96,L)
{
    const int lt   = blockIdx.x;
    const int ct   = blockIdx.y;
    const int b    = blockIdx.z;
    const int lane = threadIdx.x;
    const int mn   = lane & 15;
    const int hh   = lane >> 4;
    const int cb   = ct * 16;
    const int lb   = lt * 16;

    const float* Arow = W + (size_t)(cb + mn) * D_INNER;
    const float* Brow = x + (size_t)(b * SEQLEN + lb + mn) * D_INNER;

    v8f acc = {};
    #pragma unroll 8
    for (int k0 = 0; k0 < D_INNER; k0 += 4) {
        const int ka = k0 + 2 * hh;
        v2f a  = *(const v2f*)(Arow + ka);
        v2f bb = *(const v2f*)(Brow + ka);
        acc = __builtin_amdgcn_wmma_f32_16x16x4_f32(false, a, false, bb,
                                                    (short)0, acc, false, false);
    }

    const int col = lb + mn;
    #pragma unroll
    for (int v = 0; v < 8; ++v) {
        const int row = cb + v + hh * 8;
        float val = acc[v];
        if (row >= DT_RANK + D_STATE)   // Cs rows: add prompt
            val += prompt[(size_t)(b * SEQLEN + col) * D_STATE + (row - DT_RANK - D_STATE)];
        xdbl[(size_t)(b * NPROJ + row) * SEQLEN + col] = val;
    }
}

// -------------------------------------------------------------------------
// Kernel 2: delta[b,d,l] = softplus( sum_r dtW[d,r]*x_dbl[b,r,l] + bias[d] )
// M=2048, N=L, K=64. One wave per 16x16 tile, f32 WMMA 16x16x4.
// -------------------------------------------------------------------------
__global__ __launch_bounds__(32)
void delta_wmma_kernel(const float* __restrict__ xdbl,   // (B,96,L)
                       const float* __restrict__ dtW,    // (D_INNER,64)
                       const float* __restrict__ bias,   // (D_INNER)
                       float* __restrict__ delta)        // (B,D_INNER,L)
{
    const int lt   = blockIdx.x;
    const int dt   = blockIdx.y;
    const int b    = blockIdx.z;
    const int lane = threadIdx.x;
    const int mn   = lane & 15;
    const int hh   = lane >> 4;
    const int db   = dt * 16;
    const int lb   = lt * 16;

    const float* Arow  = dtW + (size_t)(db + mn) * DT_RANK;
    const float* Bbase = xdbl + (size_t)b * NPROJ * SEQLEN + lb + mn;

    v8f acc = {};
    #pragma unroll
    for (int k0 = 0; k0 < DT_RANK; k0 += 4) {
        const int ka = k0 + 2 * hh;
        v2f a = *(const v2f*)(Arow + ka);
        v2f bb;
        bb.x = Bbase[(size_t)ka       * SEQLEN];
        bb.y = Bbase[(size_t)(ka + 1) * SEQLEN];
        acc = __builtin_amdgcn_wmma_f32_16x16x4_f32(false, a, false, bb,
                                                    (short)0, acc, false, false);
    }

    const int col = lb + mn;
    #pragma unroll
    for (int v = 0; v < 8; ++v) {
        const int row = db + v + hh * 8;
        const float z  = acc[v] + bias[row];
        const float sp = fmaxf(z, 0.0f) + log1pf(expf(-fabsf(z)));   // softplus
        delta[(size_t)(b * D_INNER + row) * SEQLEN + col] = sp;
    }
}

// -------------------------------------------------------------------------
// Kernel 3 (pass 1): per (b,d,chunk) compute chunk decay product P[n] and
// local final state S[n] (scan from h=0). h_l = a_l*h + du_l*B_l[n],
// a_l = exp(delta_l * A[n]).  Layout: P/S[((b*D+d)*NC+c)*16 + n].
// -------------------------------------------------------------------------
__global__ __launch_bounds__(256)
void scan_pass1_kernel(const float* __restrict__ x,       // (B,L,D_INNER)
                       const float* __restrict__ xdbl,    // (B,96,L): rows 64..79 = Bs
                       const float* __restrict__ delta,   // (B,D_INNER,L)
                       const float* __restrict__ A_logs,  // (D_INNER,16)
                       float* __restrict__ P,             // (B,D_INNER,NC,16)
                       float* __restrict__ S)             // (B,D_INNER,NC,16)
{
    __shared__ float sB[D_STATE * CL];

    const int tid = threadIdx.x;
    const int d   = blockIdx.x * 256 + tid;
    const int c   = blockIdx.y;
    const int b   = blockIdx.z;
    const int l0  = c * CL;

    const float* Bsrc = xdbl + (size_t)(b * NPROJ + DT_RANK) * SEQLEN;
    for (int idx = tid; idx < D_STATE * CL; idx += 256) {
        const int n = idx >> 6;            // / CL
        const int j = idx & (CL - 1);
        sB[idx] = Bsrc[(size_t)n * SEQLEN + l0 + j];
    }
    __syncthreads();

    float A[D_STATE], h[D_STATE], p[D_STATE];
    #pragma unroll
    for (int n = 0; n < D_STATE; ++n) {
        A[n] = -expf(A_logs[(size_t)d * D_STATE + n]);
        h[n] = 0.0f;
        p[n] = 1.0f;
    }

    const float* dptr = delta + (size_t)(b * D_INNER + d) * SEQLEN + l0;
    const float* uptr = x + (size_t)(b * SEQLEN + l0) * D_INNER + d;

    for (int j = 0; j < CL; ++j) {
        const float dl = dptr[j];
        const float u  = uptr[(size_t)j * D_INNER];
        const float du = dl * u;
        #pragma unroll
        for (int n = 0; n < D_STATE; ++n) {
            const float a = expf(dl * A[n]);
            p[n] *= a;
            h[n] = a * h[n] + du * sB[n * CL + j];
        }
    }

    float* Pout = P + ((size_t)(b * D_INNER + d) * NC + c) * D_STATE;
    float* Sout = S + ((size_t)(b * D_INNER + d) * NC + c) * D_STATE;
    #pragma unroll
    for (int n = 0; n < D_STATE; ++n) {
        Pout[n] = p[n];
        Sout[n] = h[n];
    }
}

// -------------------------------------------------------------------------
// Kernel 4 (combine): per (b,d,n), fold chunks sequentially:
//   Hin[c] = H;  H = P[c]*H + S[c]
// Hin ALIASES the P buffer: each element is read (P) before written (Hin)
// by the single owning thread, so the overwrite is safe.
// -------------------------------------------------------------------------
__global__ __launch_bounds__(256)
void scan_combine_kernel(float* __restrict__ P_Hin,       // in: P, out: Hin
                         const float* __restrict__ S)
{
    const int idx  = blockIdx.x * 256 + threadIdx.x;      // (b,d,n) flat
    const int n    = idx & 15;
    const int rest = idx >> 4;
    const int d    = rest & (D_INNER - 1);
    const int b    = rest >> 11;

    const size_t base = ((size_t)(b * D_INNER + d) * NC) * D_STATE + n;
    float H = 0.0f;
    for (int c = 0; c < NC; ++c) {
        const size_t o = base + (size_t)c * D_STATE;
        const float p = P_Hin[o];
        const float s = S[o];
        P_Hin[o] = H;                 // exclusive prefix -> Hin
        H = fmaf(p, H, s);
    }
}

// -------------------------------------------------------------------------
// Kernel 5 (pass 2): rescan each chunk starting from Hin, emit outputs:
//   y_l = sum_n h_l[n]*C_l[n];  out = y + u*Ds[d]
// -------------------------------------------------------------------------
__global__ __launch_bounds__(256)
void scan_pass2_kernel(const float* __restrict__ x,       // (B,L,D_INNER)
                       const float* __restrict__ xdbl,    // rows 64..79=Bs, 80..95=Cs
                       const float* __restrict__ delta,   // (B,D_INNER,L)
                       const float* __restrict__ A_logs,  // (D_INNER,16)
                       const float* __restrict__ Ds,      // (D_INNER)
                       const float* __restrict__ Hin,     // (B,D_INNER,NC,16)
                       float* __restrict__ out)           // (B,L,D_INNER)
{
    __shared__ float sB[D_STATE * CL];
    __shared__ float sC[D_STATE * CL];

    const int tid = threadIdx.x;
    const int d   = blockIdx.x * 256 + tid;
    const int c   = blockIdx.y;
    const int b   = blockIdx.z;
    const int l0  = c * CL;

    const float* Bsrc = xdbl + (size_t)(b * NPROJ + DT_RANK) * SEQLEN;
    const float* Csrc = xdbl + (size_t)(b * NPROJ + DT_RANK + D_STATE) * SEQLEN;
    for (int idx = tid; idx < D_STATE * CL; idx += 256) {
        const int n = idx >> 6;
        const int j = idx & (CL - 1);
        sB[idx] = Bsrc[(size_t)n * SEQLEN + l0 + j];
        sC[idx] = Csrc[(size_t)n * SEQLEN + l0 + j];
    }
    __syncthreads();

    const float* hin = Hin + ((size_t)(b * D_INNER + d) * NC + c) * D_STATE;
    float A[D_STATE], h[D_STATE];
    #pragma unroll
    for (int n = 0; n < D_STATE; ++n) {
        A[n] = -expf(A_logs[(size_t)d * D_STATE + n]);
        h[n] = hin[n];
    }
    const float Dscale = Ds[d];

    const float* dptr = delta + (size_t)(b * D_INNER + d) * SEQLEN + l0;
    const float* uptr = x + (size_t)(b * SEQLEN + l0) * D_INNER + d;
    float* optr = out + (size_t)(b * SEQLEN + l0) * D_INNER + d;

    for (int j = 0; j < CL; ++j) {
        const float dl = dptr[j];
        const float u  = uptr[(size_t)j * D_INNER];
        const float du = dl * u;
        float y = 0.0f;
        #pragma unroll
        for (int n = 0; n < D_STATE; ++n) {
            const float a = expf(dl * A[n]);
            h[n] = a * h[n] + du * sB[n * CL + j];
            y += h[n] * sC[n * CL + j];
        }
        optr[(size_t)j * D_INNER] = y + u * Dscale;
    }
}

// -------------------------------------------------------------------------
extern "C" void kernel_launch(void* const* d_in, const int* in_sizes, int n_in,
                              void* d_out, int out_size, void* d_ws, size_t ws_size,
                              hipStream_t stream)
{
    const float* x      = (const float*)d_in[0];  // (B,L,D_INNER)
    const float* prompt = (const float*)d_in[1];  // (B,L,16)
    const float* xprojW = (const float*)d_in[2];  // (1,96,D_INNER)
    const float* dtW    = (const float*)d_in[3];  // (1,D_INNER,64)
    const float* dtBias = (const float*)d_in[4];  // (1,D_INNER)
    const float* A_logs = (const float*)d_in[5];  // (D_INNER,16)
    const float* Ds     = (const float*)d_in[6];  // (D_INNER)
    float* out = (float*)d_out;

    float* xdbl  = (float*)d_ws;                                  // B*96*L      (1.5 MB)
    float* delta = xdbl + (size_t)BATCH * NPROJ * SEQLEN;         // B*D*L       (32 MB)
    float* P     = delta + (size_t)BATCH * D_INNER * SEQLEN;      // B*D*NC*16   (8 MB), reused as Hin
    float* S     = P + (size_t)BATCH * D_INNER * NC * D_STATE;    // B*D*NC*16   (8 MB)

    dim3 g1(SEQLEN / 16, NPROJ / 16, BATCH);
    proj_wmma_kernel<<<g1, 32, 0, stream>>>(x, xprojW, prompt, xdbl);

    dim3 g2(SEQLEN / 16, D_INNER / 16, BATCH);
    delta_wmma_kernel<<<g2, 32, 0, stream>>>(xdbl, dtW, dtBias, delta);

    dim3 g3(D_INNER / 256, NC, BATCH);
    scan_pass1_kernel<<<g3, 256, 0, stream>>>(x, xdbl, delta, A_logs, P, S);

    const int combine_threads = BATCH * D_INNER * D_STATE;        // 65536
    scan_combine_kernel<<<combine_threads / 256, 256, 0, stream>>>(P, S);

    scan_pass2_kernel<<<g3, 256, 0, stream>>>(x, xdbl, delta, A_logs, Ds, P, out);
}